// HGPSLPool_9208409883082
// MI455X (gfx1250) — compile-verified
//
#include <hip/hip_runtime.h>
#include <hip/hip_bf16.h>

// ---------------- problem constants (match reference) ----------------
#define NB    8        // graphs
#define NPER  256      // nodes per graph
#define NN    2048     // total nodes (N)
#define FF    512      // features
#define EE    65536    // total edges
#define KK    204      // kept per graph
#define NK    1632     // pooled nodes
#define KHOP  3
#define LAMB  1.0f
#define NEG_SLOPE 0.2f

typedef __attribute__((ext_vector_type(2))) float v2f;
typedef __attribute__((ext_vector_type(8))) float v8f;

// ---------------- workspace layout (bytes) ----------------
#define OFF_A       0
#define OFF_T       (OFF_A + (size_t)NN * NN * 4)          // 16 MB
#define OFF_APOOL   (OFF_T + (size_t)NN * NN * 4)          // +16 MB
#define OFF_AGG     (OFF_APOOL + (size_t)NK * NK * 4)      // +10.65 MB
#define OFF_ODEG    (OFF_AGG + (size_t)NN * FF * 4)        // +4 MB
#define OFF_IDEG    (OFF_ODEG + (size_t)NN * 4)
#define OFF_SCORE   (OFF_IDEG + (size_t)NN * 4)
#define OFF_PARTIAL (OFF_SCORE + (size_t)NN * 4)
#define OFF_MAXT    (OFF_PARTIAL + (size_t)1024 * 4)
#define OFF_PERM    (OFF_MAXT + 256)
#define OFF_SSRC    (OFF_PERM + (size_t)NK * 4 + 128)
#define OFF_SDST    (OFF_SSRC + (size_t)NK * 4 + 128)

__device__ __forceinline__ float wave_sum32(float v) {
#pragma unroll
  for (int o = 16; o > 0; o >>= 1) v += __shfl_xor(v, o, 32);
  return v;
}

// ---------------- degrees ----------------
__global__ void degrees_kernel(const int* __restrict__ row, const int* __restrict__ col,
                               float* odeg, float* ideg) {
  for (int e = blockIdx.x * blockDim.x + threadIdx.x; e < EE; e += gridDim.x * blockDim.x) {
    atomicAdd(&odeg[row[e]], 1.0f);
    atomicAdd(&ideg[col[e]], 1.0f);
  }
}

// ---------------- message aggregation (one block per edge) ----------------
__global__ void agg_kernel(const int* __restrict__ row, const int* __restrict__ col,
                           const float* __restrict__ e_feat, const float* __restrict__ feat,
                           const float* __restrict__ odeg, float* __restrict__ agg) {
  int e = blockIdx.x;
  int r = row[e], c = col[e];
  if (r == c) return;  // remove_self_loop
  float coef = e_feat[e] * rsqrtf(fmaxf(odeg[r], 1.0f));
  const float* src = feat + (size_t)r * FF;
  float* dst = agg + (size_t)c * FF;
  for (int f = threadIdx.x; f < FF; f += blockDim.x)
    atomicAdd(&dst[f], src[f] * coef);
}

// ---------------- node info score (one wave per node) ----------------
__global__ void score_kernel(const float* __restrict__ feat, const float* __restrict__ agg,
                             const float* __restrict__ ideg, float* __restrict__ score) {
  int wave = threadIdx.x >> 5, lane = threadIdx.x & 31;
  int n = blockIdx.x * 8 + wave;
  if (n >= NN) return;
  float dn = rsqrtf(fmaxf(ideg[n], 1.0f));
  const float* fp = feat + (size_t)n * FF;
  const float* ap = agg + (size_t)n * FF;
  float s = 0.0f;
  for (int f = lane; f < FF; f += 32) s += fabsf(fp[f] - ap[f] * dn);
  s = wave_sum32(s);
  if (lane == 0) score[n] = s;
}

// ---------------- per-graph top-k via bitonic sort (desc score, idx asc) ----------------
__global__ void topk_kernel(const float* __restrict__ score, int* __restrict__ perm,
                            float* __restrict__ perm_f) {
  __shared__ float s[NPER];
  __shared__ int si[NPER];
  int b = blockIdx.x, t = threadIdx.x;
  s[t] = score[b * NPER + t];
  si[t] = t;
  __syncthreads();
  for (int k = 2; k <= NPER; k <<= 1) {
    for (int j = k >> 1; j > 0; j >>= 1) {
      int ixj = t ^ j;
      if (ixj > t) {
        float a = s[t], bb = s[ixj];
        int ia = si[t], ib = si[ixj];
        bool a_first = (a > bb) || (a == bb && ia < ib);  // descending, stable
        bool up = ((t & k) == 0);
        if (up ? !a_first : a_first) { s[t] = bb; s[ixj] = a; si[t] = ib; si[ixj] = ia; }
      }
      __syncthreads();
    }
  }
  if (t < KK) {
    int p = si[t] + b * NPER;
    perm[b * KK + t] = p;
    perm_f[b * KK + t] = (float)p;
  }
}

// ---------------- scatter COO into dense A ----------------
__global__ void scatter_adj_kernel(const int* __restrict__ row, const int* __restrict__ col,
                                   const float* __restrict__ e_feat, float* __restrict__ A) {
  for (int e = blockIdx.x * blockDim.x + threadIdx.x; e < EE; e += gridDim.x * blockDim.x)
    atomicAdd(&A[(size_t)row[e] * NN + col[e]], e_feat[e]);
}

// ---------------- WMMA f32 GEMM: C = A*B (NN x NN x NN) ----------------
// Block tile 128x64, 8 waves in 4(M) x 2(N) grid, each wave 32x32 (2x2 WMMA tiles).
// A panel in LDS rows padded to 18 floats (even -> aligned b64 k-pairs).
// B panel stored k-pair interleaved: (k,n) -> Bs[k>>1][n*2 + (k&1)], row stride 160
// floats (+32 pad) so the two lane-halves hit disjoint bank halves.
#define BM 128
#define BN 64
#define BK 16
#define ASTRIDE (BK + 2)          // 18 floats
#define BSTRIDE (BN * 2 + 32)     // 160 floats
__global__ void __launch_bounds__(256) wmma_gemm_kernel(const float* __restrict__ Am,
                                                        const float* __restrict__ Bm,
                                                        float* __restrict__ Cm) {
  __shared__ float As[BM * ASTRIDE];
  __shared__ float Bs[(BK / 2) * BSTRIDE];
  int bm = blockIdx.y * BM;
  int bn = blockIdx.x * BN;
  int t = threadIdx.x;
  int wave = t >> 5, lane = t & 31;
  int hv = lane >> 4, l16 = lane & 15;
  int wm = (wave & 3) * 32;        // wave row offset in block tile
  int wn = (wave >> 2) * 32;       // wave col offset in block tile
  // A panel load map: 128 rows x 4 float4 -> 2 float4 per thread
  int ar = t >> 2, ac = (t & 3) * 4;
  // B panel load map: 16 rows x 16 float4 -> 1 float4 per thread
  int br = t >> 4, bc = (t & 15) * 4;
  v8f acc00 = {}, acc01 = {}, acc10 = {}, acc11 = {};
  const float* AsR0 = &As[(wm + l16) * ASTRIDE];
  const float* AsR1 = &As[(wm + 16 + l16) * ASTRIDE];
  for (int k0 = 0; k0 < NN; k0 += BK) {
    const float4 av0 = *(const float4*)(Am + (size_t)(bm + ar) * NN + k0 + ac);
    const float4 av1 = *(const float4*)(Am + (size_t)(bm + 64 + ar) * NN + k0 + ac);
    const float4 bv  = *(const float4*)(Bm + (size_t)(k0 + br) * NN + bn + bc);
    if (k0 + BK < NN) {  // prefetch next K panel -> global_prefetch_b8
      __builtin_prefetch(Am + (size_t)(bm + ar) * NN + k0 + BK + ac, 0, 1);
      __builtin_prefetch(Am + (size_t)(bm + 64 + ar) * NN + k0 + BK + ac, 0, 1);
      __builtin_prefetch(Bm + (size_t)(k0 + BK + br) * NN + bn + bc, 0, 1);
    }
    __syncthreads();  // previous stage LDS reads done
    {
      float* a0 = &As[ar * ASTRIDE + ac];
      a0[0] = av0.x; a0[1] = av0.y; a0[2] = av0.z; a0[3] = av0.w;
      float* a1 = &As[(ar + 64) * ASTRIDE + ac];
      a1[0] = av1.x; a1[1] = av1.y; a1[2] = av1.z; a1[3] = av1.w;
      float* bp = &Bs[(br >> 1) * BSTRIDE + (br & 1)];
      bp[(bc + 0) * 2] = bv.x;
      bp[(bc + 1) * 2] = bv.y;
      bp[(bc + 2) * 2] = bv.z;
      bp[(bc + 3) * 2] = bv.w;
    }
    __syncthreads();
#pragma unroll
    for (int kk = 0; kk < BK / 4; ++kk) {
      int kb = kk * 4 + 2 * hv;  // even; ISA A-frag: VGPR v holds K = v + 2*(lane>=16)
      v2f a0 = *(const v2f*)(AsR0 + kb);
      v2f a1 = *(const v2f*)(AsR1 + kb);
      const float* bq = &Bs[(kb >> 1) * BSTRIDE];
      v2f b0 = *(const v2f*)(bq + (wn + l16) * 2);
      v2f b1 = *(const v2f*)(bq + (wn + 16 + l16) * 2);
      acc00 = __builtin_amdgcn_wmma_f32_16x16x4_f32(false, a0, false, b0, (short)0, acc00, false, false);
      acc01 = __builtin_amdgcn_wmma_f32_16x16x4_f32(false, a0, false, b1, (short)0, acc01, false, false);
      acc10 = __builtin_amdgcn_wmma_f32_16x16x4_f32(false, a1, false, b0, (short)0, acc10, false, false);
      acc11 = __builtin_amdgcn_wmma_f32_16x16x4_f32(false, a1, false, b1, (short)0, acc11, false, false);
    }
  }
  // C/D layout: VGPR r -> M = r + 8*(lane>=16), N = lane&15
#pragma unroll
  for (int r = 0; r < 8; ++r) {
    int m0 = bm + wm + r + 8 * hv;
    int m1 = m0 + 16;
    Cm[(size_t)m0 * NN + bn + wn + l16]      = acc00[r];
    Cm[(size_t)m0 * NN + bn + wn + 16 + l16] = acc01[r];
    Cm[(size_t)m1 * NN + bn + wn + l16]      = acc10[r];
    Cm[(size_t)m1 * NN + bn + wn + 16 + l16] = acc11[r];
  }
}

// ---------------- max reduction over T ----------------
__global__ void reduce_max_partial(const float* __restrict__ Tm, float* __restrict__ partial, int n) {
  float m = -3.4e38f;
  for (int i = blockIdx.x * blockDim.x + threadIdx.x; i < n; i += gridDim.x * blockDim.x)
    m = fmaxf(m, Tm[i]);
#pragma unroll
  for (int o = 16; o > 0; o >>= 1) m = fmaxf(m, __shfl_xor(m, o, 32));
  __shared__ float s[8];
  if ((threadIdx.x & 31) == 0) s[threadIdx.x >> 5] = m;
  __syncthreads();
  if (threadIdx.x == 0) {
    float r = s[0];
#pragma unroll
    for (int w = 1; w < 8; ++w) r = fmaxf(r, s[w]);
    partial[blockIdx.x] = r;
  }
}

__global__ void reduce_max_final(const float* __restrict__ partial, int n, float* __restrict__ maxT) {
  float m = -3.4e38f;
  for (int i = threadIdx.x; i < n; i += blockDim.x) m = fmaxf(m, partial[i]);
#pragma unroll
  for (int o = 16; o > 0; o >>= 1) m = fmaxf(m, __shfl_xor(m, o, 32));
  __shared__ float s[8];
  if ((threadIdx.x & 31) == 0) s[threadIdx.x >> 5] = m;
  __syncthreads();
  if (threadIdx.x == 0) {
    float r = s[0];
#pragma unroll
    for (int w = 1; w < 8; ++w) r = fmaxf(r, s[w]);
    maxT[0] = r;
  }
}

// ---------------- A = T * (1e-5/maxT) + A ----------------
__global__ void axpy_kernel(float* __restrict__ A, const float* __restrict__ Tm,
                            const float* __restrict__ maxT, int n) {
  float sc = 1e-5f / maxT[0];
  for (int i = blockIdx.x * blockDim.x + threadIdx.x; i < n; i += gridDim.x * blockDim.x)
    A[i] = fmaf(Tm[i], sc, A[i]);
}

// ---------------- feat_p = feat[perm] ----------------
__global__ void gather_feat_kernel(const float* __restrict__ feat, const int* __restrict__ perm,
                                   float* __restrict__ feat_p) {
  int i = blockIdx.x;
  int pi = perm[i];
  const float4* src = (const float4*)(feat + (size_t)pi * FF);
  float4* dst = (float4*)(feat_p + (size_t)i * FF);
  for (int j = threadIdx.x; j < FF / 4; j += blockDim.x) dst[j] = src[j];
}

// ---------------- A_pool = A[perm][:,perm] ----------------
__global__ void pool_adj_kernel(const float* __restrict__ A, const int* __restrict__ perm,
                                float* __restrict__ A_pool) {
  __shared__ int p[NK];
  for (int i = threadIdx.x; i < NK; i += blockDim.x) p[i] = perm[i];
  __syncthreads();
  int i = blockIdx.x;
  const float* Ar = A + (size_t)p[i] * NN;
  float* Dr = A_pool + (size_t)i * NK;
  for (int j = threadIdx.x; j < NK; j += blockDim.x) Dr[j] = Ar[p[j]];
}

// ---------------- attention projections (one wave per pooled node) ----------------
__global__ void attn_kernel(const float* __restrict__ feat_p, const float* __restrict__ att,
                            float* __restrict__ s_src, float* __restrict__ s_dst) {
  int wave = threadIdx.x >> 5, lane = threadIdx.x & 31;
  int i = blockIdx.x * 8 + wave;
  if (i >= NK) return;
  const float* fp = feat_p + (size_t)i * FF;
  float a = 0.0f, b = 0.0f;
  for (int f = lane; f < FF; f += 32) {
    float v = fp[f];
    a += v * att[f];
    b += v * att[FF + f];
  }
  a = wave_sum32(a);
  b = wave_sum32(b);
  if (lane == 0) { s_src[i] = a; s_dst[i] = b; }
}

// ---------------- sparsemax per destination column ----------------
#define SORT_N 2048
__device__ __forceinline__ float masked_logit(const float* A_pool, const float* s_src,
                                              float sdj, int i, int j, bool* mask_out) {
  float ap = A_pool[(size_t)i * NK + j];
  bool m = (ap != 0.0f) || (i == j);
  *mask_out = m;
  if (!m) return -1e9f;
  float raw = s_src[i] + sdj;
  float lr = raw > 0.0f ? raw : NEG_SLOPE * raw;
  return lr + LAMB * ap;
}

__global__ void __launch_bounds__(256) sparsemax_kernel(const float* __restrict__ s_src,
                                                        const float* __restrict__ s_dst,
                                                        const float* __restrict__ A_pool,
                                                        float* __restrict__ weights) {
  __shared__ float zs[SORT_N];
  __shared__ float cs[SORT_N];
  __shared__ float red[256];
  __shared__ int cnt_s;
  __shared__ float tau_s;
  int j = blockIdx.x;
  int t = threadIdx.x;
  float sdj = s_dst[j];
  for (int i = t; i < SORT_N; i += 256) {
    float z = -3.0e9f;  // padding: sorts last, never enters support
    if (i < NK) { bool m; z = masked_logit(A_pool, s_src, sdj, i, j, &m); }
    zs[i] = z;
  }
  __syncthreads();
  // bitonic sort, descending
  for (int k = 2; k <= SORT_N; k <<= 1) {
    for (int jj = k >> 1; jj > 0; jj >>= 1) {
      for (int i = t; i < SORT_N; i += 256) {
        int ixj = i ^ jj;
        if (ixj > i) {
          float a = zs[i], b = zs[ixj];
          bool up = ((i & k) == 0);
          if (up ? (a < b) : (a > b)) { zs[i] = b; zs[ixj] = a; }
        }
      }
      __syncthreads();
    }
  }
  // inclusive cumsum: 8 contiguous values per thread + scan of chunk sums
  int base = t * 8;
  float local = 0.0f;
#pragma unroll
  for (int q = 0; q < 8; ++q) local += zs[base + q];
  red[t] = local;
  __syncthreads();
  if (t == 0) {
    float acc = 0.0f;
    for (int q = 0; q < 256; ++q) { float v = red[q]; red[q] = acc; acc += v; }
    cnt_s = 0;
  }
  __syncthreads();
  float acc = red[t];
#pragma unroll
  for (int q = 0; q < 8; ++q) { acc += zs[base + q]; cs[base + q] = acc; }
  __syncthreads();
  // support size: cond true only on a prefix
  int c = 0;
  for (int i = t; i < SORT_N; i += 256)
    if (1.0f + (float)(i + 1) * zs[i] > cs[i]) c++;
  atomicAdd(&cnt_s, c);
  __syncthreads();
  if (t == 0) {
    int ks = cnt_s < 1 ? 1 : cnt_s;
    tau_s = (cs[ks - 1] - 1.0f) / (float)ks;
  }
  __syncthreads();
  float tau = tau_s;
  for (int i = t; i < NK; i += 256) {
    bool m;
    float z = masked_logit(A_pool, s_src, sdj, i, j, &m);
    weights[(size_t)i * NK + j] = m ? fmaxf(z - tau, 0.0f) : 0.0f;
  }
}

// ---------------- host orchestration ----------------
extern "C" void kernel_launch(void* const* d_in, const int* in_sizes, int n_in,
                              void* d_out, int out_size, void* d_ws, size_t ws_size,
                              hipStream_t stream) {
  const float* feat   = (const float*)d_in[0];
  const float* e_feat = (const float*)d_in[1];
  const float* att    = (const float*)d_in[2];
  const int*   row    = (const int*)d_in[3];
  const int*   col    = (const int*)d_in[4];

  char* ws = (char*)d_ws;
  float* A       = (float*)(ws + OFF_A);
  float* Tm      = (float*)(ws + OFF_T);
  float* A_pool  = (float*)(ws + OFF_APOOL);
  float* agg     = (float*)(ws + OFF_AGG);
  float* odeg    = (float*)(ws + OFF_ODEG);
  float* ideg    = (float*)(ws + OFF_IDEG);
  float* score   = (float*)(ws + OFF_SCORE);
  float* partial = (float*)(ws + OFF_PARTIAL);
  float* maxT    = (float*)(ws + OFF_MAXT);
  int*   perm    = (int*)(ws + OFF_PERM);
  float* s_src   = (float*)(ws + OFF_SSRC);
  float* s_dst   = (float*)(ws + OFF_SDST);

  float* feat_p  = (float*)d_out;                       // [NK, FF]
  float* weights = feat_p + (size_t)NK * FF;            // [NK, NK]
  float* perm_f  = weights + (size_t)NK * NK;           // [NK] as float

  hipMemsetAsync(A, 0, (size_t)NN * NN * 4, stream);
  hipMemsetAsync(agg, 0, (size_t)NN * FF * 4, stream);
  hipMemsetAsync(odeg, 0, (size_t)NN * 4, stream);
  hipMemsetAsync(ideg, 0, (size_t)NN * 4, stream);

  degrees_kernel<<<64, 256, 0, stream>>>(row, col, odeg, ideg);
  agg_kernel<<<EE, 256, 0, stream>>>(row, col, e_feat, feat, odeg, agg);
  score_kernel<<<NN / 8, 256, 0, stream>>>(feat, agg, ideg, score);
  topk_kernel<<<NB, NPER, 0, stream>>>(score, perm, perm_f);
  scatter_adj_kernel<<<64, 256, 0, stream>>>(row, col, e_feat, A);

  for (int it = 0; it < KHOP; ++it) {
    wmma_gemm_kernel<<<dim3(NN / BN, NN / BM), 256, 0, stream>>>(A, A, Tm);
    reduce_max_partial<<<1024, 256, 0, stream>>>(Tm, partial, NN * NN);
    reduce_max_final<<<1, 256, 0, stream>>>(partial, 1024, maxT);
    axpy_kernel<<<2048, 256, 0, stream>>>(A, Tm, maxT, NN * NN);
  }

  gather_feat_kernel<<<NK, 128, 0, stream>>>(feat, perm, feat_p);
  pool_adj_kernel<<<NK, 256, 0, stream>>>(A, perm, A_pool);
  attn_kernel<<<NK / 8, 256, 0, stream>>>(feat_p, att, s_src, s_dst);
  sparsemax_kernel<<<NK, 256, 0, stream>>>(s_src, s_dst, A_pool, weights);
}